// DeformedAgent_46471546143007
// MI455X (gfx1250) — compile-verified
//
#include <hip/hip_runtime.h>

typedef __attribute__((ext_vector_type(2))) float v2f;
typedef __attribute__((ext_vector_type(8))) float v8f;

#define B_    8
#define L_    4096
#define C_    1024
#define LK_   64
#define KCH   64      // K chunk staged in LDS per iteration
#define APAD  68      // padded LDS row stride in floats (bank-conflict-free)
#define COW   128     // output channels covered per workgroup (8 waves x 16)

// Y[b, j, co] = sum_c X[b, j, c] * wq[co, c] + bq[co]
// X[b, j, c]  = tokens[b, 1 + ...] at flat (j*16 + (c>>6))*4096 + 65*(c&63)
// (the conv/LN/GELU/tanh offset path provably yields idx == k and is eliminated)
__global__ __launch_bounds__(256)
void deformed_agent_gemm(const float* __restrict__ tokens,
                         const float* __restrict__ wq,
                         const float* __restrict__ bq,
                         float* __restrict__ out)
{
    __shared__ float As[16  * APAD];   // gathered X tile: 16 rows x 64 k
    __shared__ float Bs[COW * APAD];   // wq tile: 128 co x 64 k

    const int t   = threadIdx.x;
    const int w   = t >> 5;                 // wave id 0..7
    const int ln  = t & 31;                 // lane within wave32
    const int b   = blockIdx.z;             // batch
    const int Jb  = blockIdx.y << 4;        // j-tile base (16 rows)
    const int CO0 = blockIdx.x * COW;       // co chunk base
    const int N0  = CO0 + (w << 4);         // this wave's 16 output channels

    // tokens[:, CLS:, :] slice base for this batch
    const float* tok = tokens + (size_t)b * (size_t)((L_ + 1) * C_) + C_;

    // WMMA f32 16x16x4 lane decomposition:
    //   A: lane l holds A[m][kb], A[m][kb+1]   (m = l&15, kb = (l>>4)*2)
    //   B: lane l holds B[kb][n], B[kb+1][n]   (n = l&15)
    const int mn = ln & 15;
    const int kb = (ln >> 4) << 1;

    v8f acc = {};

    for (int kc0 = 0; kc0 < C_; kc0 += KCH) {
        const int cblk = kc0 >> 6;   // constant (c>>6) contribution this chunk

        // ---- stage A tile: strided gather of X[Jb..Jb+15][kc0..kc0+63] ----
        #pragma unroll
        for (int i = 0; i < 4; ++i) {
            int flat = t + (i << 8);            // 0..1023
            int r    = flat >> 6;               // row 0..15
            int kl   = flat & 63;               // k within chunk
            int j    = Jb + r;
            As[r * APAD + kl] =
                tok[(size_t)(j * 16 + cblk) * L_ + 65 * kl];
        }
        // ---- stage B tile: wq[CO0..CO0+127][kc0..kc0+63], coalesced b128 ----
        #pragma unroll
        for (int i = 0; i < 8; ++i) {
            int flat = t + (i << 8);            // 0..2047
            int row  = flat >> 4;               // co row 0..127
            int c4   = flat & 15;               // float4 column
            const float4 v =
                *(const float4*)(wq + (size_t)(CO0 + row) * C_ + kc0 + (c4 << 2));
            *(float4*)(Bs + row * APAD + (c4 << 2)) = v;
        }
        __syncthreads();

        // ---- 16 WMMA steps over this K chunk ----
        const float* arow = As + mn * APAD + kb;
        const float* brow = Bs + ((w << 4) + mn) * APAD + kb;
        #pragma unroll
        for (int k4 = 0; k4 < KCH; k4 += 4) {
            v2f a  = *(const v2f*)(arow + k4);
            v2f bb = *(const v2f*)(brow + k4);
            acc = __builtin_amdgcn_wmma_f32_16x16x4_f32(
                      false, a, false, bb, (short)0, acc, false, false);
        }
        __syncthreads();
    }

    // ---- epilogue: add bias, store (output flat layout == Y[b,j,co] flat) ----
    const int   co    = N0 + mn;
    const float bias  = bq[co];
    const int   mbase = (ln >> 4) << 3;     // 0 or 8
    float* outb = out + (size_t)b * (size_t)(LK_ * C_);
    #pragma unroll
    for (int r = 0; r < 8; ++r) {
        int j = Jb + mbase + r;
        outb[(size_t)j * C_ + co] = acc[r] + bias;
    }
}

extern "C" void kernel_launch(void* const* d_in, const int* in_sizes, int n_in,
                              void* d_out, int out_size, void* d_ws, size_t ws_size,
                              hipStream_t stream) {
    // setup_inputs order: q(0), tokens(1), w_dw(2), b_dw(3), ln_w(4), ln_b(5),
    //                     w_off(6), wq(7), bq(8)
    const float* tokens = (const float*)d_in[1];
    const float* wq     = (const float*)d_in[7];
    const float* bq     = (const float*)d_in[8];
    float* out = (float*)d_out;

    dim3 grid(C_ / COW, LK_ / 16, B_);   // (8, 4, 8) = 256 workgroups
    deformed_agent_gemm<<<grid, dim3(256, 1, 1), 0, stream>>>(tokens, wq, bq, out);
}